// Score_429496730110
// MI455X (gfx1250) — compile-verified
//
#include <hip/hip_runtime.h>
#include <hip/hip_bf16.h>

// Problem constants (match reference)
#define N_SAMPLES 512
#define NOBJ      16
#define M_PTS     8192
#define H_ROI     128
#define W_ROI     128
#define HW        (H_ROI * W_ROI)

typedef float v2f __attribute__((ext_vector_type(2)));
typedef float v8f __attribute__((ext_vector_type(8)));

// ---------------------------------------------------------------------------
// wave32 reduction (CDNA5 is wave32-only)
// ---------------------------------------------------------------------------
__device__ __forceinline__ float wave_reduce_sum(float v) {
    v += __shfl_xor(v, 16, 32);
    v += __shfl_xor(v, 8, 32);
    v += __shfl_xor(v, 4, 32);
    v += __shfl_xor(v, 2, 32);
    v += __shfl_xor(v, 1, 32);
    return v;
}

// ---------------------------------------------------------------------------
// Build per-sample A matrix for V_WMMA_F32_16X16X4_F32 (16x4, f32).
// A layout (ISA 7.12.2): lanes 0-15 hold K=0 (VGPR0) / K=1 (VGPR1) of row
// M = lane; lanes 16-31 hold K=2 / K=3 of row M = lane-16.
//
// Rows (K=0..2 rotation, K=3 translation), projection-folded:
//   row0 = fx*(Rp[0,:] | tp.x)   row3 = fx*(Rg[0,:] | tg.x)
//   row1 = fy*(Rp[1,:] | tp.y)   row4 = fy*(Rg[1,:] | tg.y)
//   row2 =    (Rp[2,:] | tp.z)   row5 =    (Rg[2,:] | tg.z)
//   rows 6..15 = 0
// ---------------------------------------------------------------------------
__device__ __forceinline__ v2f build_A(int n, int lane,
                                       const float* __restrict__ Rp,
                                       const float* __restrict__ tp,
                                       const float* __restrict__ Rg,
                                       const float* __restrict__ tg,
                                       float fx, float fy) {
    int  m       = lane & 15;
    bool valid   = (m < 6);
    int  mm      = valid ? m : 0;
    bool usePred = (mm < 3);
    int  r       = usePred ? mm : (mm - 3);
    const float* R = usePred ? (Rp + n * 9) : (Rg + n * 9);
    const float* t = usePred ? (tp + n * 3) : (tg + n * 3);
    float s  = (r == 0) ? fx : ((r == 1) ? fy : 1.0f);
    int   k0 = (lane < 16) ? 0 : 2;           // K index in VGPR0 (always < 3)
    float e0 = R[r * 3 + k0];
    float e1 = (lane < 16) ? R[r * 3 + 1]     // K=1: rotation element
                           : t[r];            // K=3: translation component
    v2f a;
    a.x = valid ? s * e0 : 0.0f;
    a.y = valid ? s * e1 : 0.0f;
    return a;
}

// ---------------------------------------------------------------------------
// PV: masked mean of projected distance over ROI pixels.
// One block (8 waves) per sample; each wave streams a contiguous 2048-pixel
// chunk, 16 pixels per WMMA. Bandwidth-bound: ~100 MB of coords total
// (~4.5 us at 23.3 TB/s). Epilogue uses raw v_rcp_f32/v_sqrt_f32 so the
// per-pixel VALU work stays well under the HBM time.
// ---------------------------------------------------------------------------
__global__ __launch_bounds__(256) void pv_kernel(
        const float* __restrict__ camK,
        const float* __restrict__ Rg, const float* __restrict__ tg,
        const float* __restrict__ Rp, const float* __restrict__ tp,
        const float* __restrict__ coord,        // [N,3,H,W] planar
        const unsigned char* __restrict__ mask, // [N,1,H,W] bool
        float* __restrict__ pv_out) {
    const int n    = blockIdx.x;
    const int lane = threadIdx.x & 31;
    const int wid  = threadIdx.x >> 5;

    const float fx = camK[n * 9 + 0];
    const float fy = camK[n * 9 + 4];
    const v2f   a  = build_A(n, lane, Rp, tp, Rg, tg, fx, fy);

    const float*         C3 = coord + (size_t)n * 3 * HW;
    const unsigned char* Mk = mask  + (size_t)n * HW;

    const int planeOff = (lane < 16) ? 0 : 2 * HW;  // x-plane vs z-plane
    const int idx      = lane & 15;
    const int base     = wid * (HW / 8);            // 2048 pixels per wave

    float sum = 0.0f, cnt = 0.0f;
    for (int g = 0; g < HW / 8; g += 16) {
        const int pix = base + g + idx;
        // speculative streaming prefetch (global_prefetch_b8)
        __builtin_prefetch(&C3[planeOff + pix + 512], 0, 1);

        const float xz = C3[planeOff + pix];        // x (lanes<16) or z
        const float yv = C3[HW + pix];              // y plane
        v2f b;
        b.x = xz;
        b.y = (lane < 16) ? yv : 1.0f;              // K=1 -> y ; K=3 -> 1
        const float mb = (float)Mk[pix];

        v8f c = {};
        c = __builtin_amdgcn_wmma_f32_16x16x4_f32(false, a, false, b,
                                                  (short)0, c, false, false);
        // lane L<16 holds: c0=fx*xp c1=fy*yp c2=zp c3=fx*xg c4=fy*yg c5=zg
        const float rp = __builtin_amdgcn_rcpf(c[2]);
        const float rg = __builtin_amdgcn_rcpf(c[5]);
        const float du = c[0] * rp - c[3] * rg;
        const float dv = c[1] * rp - c[4] * rg;
        const float d  = __builtin_amdgcn_sqrtf(du * du + dv * dv);

        const float dm = (lane < 16) ? d  : 0.0f;   // kill garbage lanes
        const float mk = (lane < 16) ? mb : 0.0f;
        sum += dm * mk;
        cnt += mk;
    }

    sum = wave_reduce_sum(sum);
    cnt = wave_reduce_sum(cnt);

    __shared__ float ssum[8], scnt[8];
    if (lane == 0) { ssum[wid] = sum; scnt[wid] = cnt; }
    __syncthreads();
    if (threadIdx.x == 0) {
        float S = 0.0f, C = 0.0f;
        #pragma unroll
        for (int i = 0; i < 8; ++i) { S += ssum[i]; C += scnt[i]; }
        pv_out[n] = S / fmaxf(C, 1.0f);
    }
}

// ---------------------------------------------------------------------------
// AD + PJ over mesh points (gathered by obj_id; 16 objects -> L2 resident).
// Same WMMA mapping; AD displacement recovered lane-locally by un-folding
// fx/fy:  ddx=(c0-c3)/fx  ddy=(c1-c4)/fy  ddz=c2-c5.
// ---------------------------------------------------------------------------
__global__ __launch_bounds__(256) void mesh_kernel(
        const int*   __restrict__ obj_id,
        const float* __restrict__ camK,
        const float* __restrict__ Rg, const float* __restrict__ tg,
        const float* __restrict__ Rp, const float* __restrict__ tp,
        const float* __restrict__ mesh,   // [NOBJ, M, 3] interleaved
        const float* __restrict__ diam,   // [NOBJ]
        float* __restrict__ ad_out, float* __restrict__ pj_out) {
    const int n    = blockIdx.x;
    const int lane = threadIdx.x & 31;
    const int wid  = threadIdx.x >> 5;

    const float fx = camK[n * 9 + 0];
    const float fy = camK[n * 9 + 4];
    const float inv_fx = __builtin_amdgcn_rcpf(fx);
    const float inv_fy = __builtin_amdgcn_rcpf(fy);
    const v2f a = build_A(n, lane, Rp, tp, Rg, tg, fx, fy);

    const int obj = obj_id[n];
    const float* P = mesh + (size_t)obj * M_PTS * 3;

    const int idx   = lane & 15;
    const int comp0 = (lane < 16) ? 0 : 2;   // x or z component
    const int base  = wid * (M_PTS / 8);     // 1024 points per wave

    float sad = 0.0f, spj = 0.0f;
    for (int g = 0; g < M_PTS / 8; g += 16) {
        const int m = base + g + idx;
        const float xz = P[3 * m + comp0];
        const float yv = P[3 * m + 1];
        v2f b;
        b.x = xz;
        b.y = (lane < 16) ? yv : 1.0f;

        v8f c = {};
        c = __builtin_amdgcn_wmma_f32_16x16x4_f32(false, a, false, b,
                                                  (short)0, c, false, false);
        // ADD displacement (un-fold intrinsics)
        const float ddx = (c[0] - c[3]) * inv_fx;
        const float ddy = (c[1] - c[4]) * inv_fy;
        const float ddz =  c[2] - c[5];
        const float ad  = __builtin_amdgcn_sqrtf(ddx * ddx + ddy * ddy + ddz * ddz);
        // projected distance
        const float rp = __builtin_amdgcn_rcpf(c[2]);
        const float rg = __builtin_amdgcn_rcpf(c[5]);
        const float du = c[0] * rp - c[3] * rg;
        const float dv = c[1] * rp - c[4] * rg;
        const float pj = __builtin_amdgcn_sqrtf(du * du + dv * dv);

        sad += (lane < 16) ? ad : 0.0f;
        spj += (lane < 16) ? pj : 0.0f;
    }

    sad = wave_reduce_sum(sad);
    spj = wave_reduce_sum(spj);

    __shared__ float s_ad[8], s_pj[8];
    if (lane == 0) { s_ad[wid] = sad; s_pj[wid] = spj; }
    __syncthreads();
    if (threadIdx.x == 0) {
        float A_ = 0.0f, Pj = 0.0f;
        #pragma unroll
        for (int i = 0; i < 8; ++i) { A_ += s_ad[i]; Pj += s_pj[i]; }
        const float invM = 1.0f / (float)M_PTS;
        ad_out[n] = A_ * invM / diam[obj];
        pj_out[n] = Pj * invM;
    }
}

// ---------------------------------------------------------------------------
// RE (deg) + TE: trivial per-sample scalar work.
// trace(Rp @ Rg^T) = sum_ij Rp[i][j]*Rg[i][j]
// ---------------------------------------------------------------------------
__global__ void rete_kernel(const float* __restrict__ Rg,
                            const float* __restrict__ tg,
                            const float* __restrict__ Rp,
                            const float* __restrict__ tp,
                            float* __restrict__ re_out,
                            float* __restrict__ te_out) {
    const int n = blockIdx.x * blockDim.x + threadIdx.x;
    if (n >= N_SAMPLES) return;
    const float* A = Rp + n * 9;
    const float* B = Rg + n * 9;
    float tr = 0.0f;
    #pragma unroll
    for (int i = 0; i < 9; ++i) tr += A[i] * B[i];
    tr = fminf(fmaxf(tr, -1.0f), 3.0f);
    re_out[n] = acosf((tr - 1.0f) * 0.5f) * (180.0f / 3.14159265358979f);

    const float dx = tp[n * 3 + 0] - tg[n * 3 + 0];
    const float dy = tp[n * 3 + 1] - tg[n * 3 + 1];
    const float dz = tp[n * 3 + 2] - tg[n * 3 + 2];
    te_out[n] = sqrtf(dx * dx + dy * dy + dz * dz) * 100.0f;
}

// ---------------------------------------------------------------------------
// Launch
// in:  0 obj_id(i32) 1 cam_K 2 gt_R 3 gt_t 4 pred_R 5 pred_t
//      6 gt_coord_3d_roi 7 gt_mask_vis_roi(bool) 8 mesh_points 9 diameters
// out: [re|te|ad|pj|pv], each N_SAMPLES floats
// ---------------------------------------------------------------------------
extern "C" void kernel_launch(void* const* d_in, const int* in_sizes, int n_in,
                              void* d_out, int out_size, void* d_ws, size_t ws_size,
                              hipStream_t stream) {
    const int*           obj_id = (const int*)d_in[0];
    const float*         camK   = (const float*)d_in[1];
    const float*         Rg     = (const float*)d_in[2];
    const float*         tg     = (const float*)d_in[3];
    const float*         Rp     = (const float*)d_in[4];
    const float*         tp     = (const float*)d_in[5];
    const float*         coord  = (const float*)d_in[6];
    const unsigned char* mask   = (const unsigned char*)d_in[7];
    const float*         mesh   = (const float*)d_in[8];
    const float*         diam   = (const float*)d_in[9];

    float* out = (float*)d_out;
    float* re  = out + 0 * N_SAMPLES;
    float* te  = out + 1 * N_SAMPLES;
    float* ad  = out + 2 * N_SAMPLES;
    float* pj  = out + 3 * N_SAMPLES;
    float* pv  = out + 4 * N_SAMPLES;

    rete_kernel<<<(N_SAMPLES + 255) / 256, 256, 0, stream>>>(Rg, tg, Rp, tp, re, te);
    mesh_kernel<<<N_SAMPLES, 256, 0, stream>>>(obj_id, camK, Rg, tg, Rp, tp,
                                               mesh, diam, ad, pj);
    pv_kernel<<<N_SAMPLES, 256, 0, stream>>>(camK, Rg, tg, Rp, tp, coord, mask, pv);
}